// Gemma4Router_19275813224546
// MI455X (gfx1250) — compile-verified
//
#include <hip/hip_runtime.h>

typedef float v2f __attribute__((ext_vector_type(2)));
typedef float v8f __attribute__((ext_vector_type(8)));

#define H_DIM   2048
#define E_DIM   64
#define T_TOK   16384
#define KCHUNK  64
#define NWAVES  8
#define TPW     16              // tokens per wave
#define TPB     128             // tokens per block
#define XPITCH  68              // padded x/logits row pitch (floats)
#define NKSTEP  (KCHUNK / 4)    // 16 k-steps per chunk
#define WB_SZ   (NKSTEP * 4 * 32 * 2)   // 4096 floats, B-layout W chunk

// ---- kernel 1: fold per_dim_scale into W:  Wp[h,e] = pds[h] * W[h,e] ----
__global__ void fold_scale_kernel(const float* __restrict__ W,
                                  const float* __restrict__ pds,
                                  float* __restrict__ Wp) {
    int idx = blockIdx.x * blockDim.x + threadIdx.x;   // 0 .. H*E-1
    int h = idx >> 6;                                  // E_DIM == 64
    Wp[idx] = W[idx] * pds[h];
}

// ---- kernel 2: fused RMSNorm-scale + GEMM (WMMA f32 16x16x4) + top-2 router ----
__global__ __launch_bounds__(256)
void router_kernel(const float* __restrict__ x,
                   const float* __restrict__ Wp,
                   float* __restrict__ out) {
    // W chunk stored directly in WMMA-B VGPR layout:
    //   wB[((kk*4 + n)*32 + lane)*2 + j] = W'[kk*4 + 2*(lane>>4) + j][n*16 + (lane&15)]
    __shared__ float lds_wB[WB_SZ];                       // 16.0 KB (shared W' chunk)
    __shared__ float lds_x[NWAVES][TPW * XPITCH];         // 34.0 KB (per-wave x tile / logits)
    __shared__ float lds_sq[256];                         //  1.0 KB (sum-of-squares partials)

    const int tid  = threadIdx.x;
    const int wave = tid >> 5;
    const int lane = tid & 31;
    const int half = lane >> 4;        // 0: lanes 0-15, 1: lanes 16-31
    const int nl   = lane & 15;

    const int waveTok = blockIdx.x * TPB + wave * TPW;

    // x staging role: lane pair (2t, 2t+1) covers token t's 64-float chunk row
    const int mytok = lane >> 1;
    const int sub   = lane & 1;
    const float* xrow = x + (size_t)(waveTok + mytok) * H_DIM + sub * 32;
    float*       xl   = &lds_x[wave][mytok * XPITCH + sub * 32];

    float sq = 0.f;
    v8f acc0 = {}, acc1 = {}, acc2 = {}, acc3 = {};

    for (int kc = 0; kc < H_DIM; kc += KCHUNK) {
        // --- stage W' chunk, swizzled into B layout (4 x float4 / thread) ---
        #pragma unroll
        for (int s2 = 0; s2 < 4; ++s2) {
            int g   = (tid + s2 * 256) * 4;            // float offset in chunk [0,4096)
            int row = g >> 6, col = g & 63;            // row = K, col = expert
            float4 wv = *(const float4*)(Wp + (size_t)kc * E_DIM + g);
            int kk = row >> 2, h2 = (row >> 1) & 1, j = row & 1;
            int n  = col >> 4, nl0 = col & 15;         // 4 cols stay inside one 16-wide tile
            int t0 = ((kk * 4 + n) * 32 + h2 * 16 + nl0) * 2 + j;
            lds_wB[t0 + 0] = wv.x;
            lds_wB[t0 + 2] = wv.y;
            lds_wB[t0 + 4] = wv.z;
            lds_wB[t0 + 6] = wv.w;
        }
        // --- stage x tile [16 x KCHUNK] (wave-private), accumulate sum(x^2) ---
        #pragma unroll
        for (int j = 0; j < 8; ++j) {
            float4 v = *(const float4*)(xrow + kc + j * 4);
            sq += v.x * v.x + v.y * v.y + v.z * v.z + v.w * v.w;
            *(float4*)(xl + j * 4) = v;
        }
        __syncthreads();

        // --- WMMA: D(16x64) += A(16x4) x B(4x64) over the chunk ---
        const float* xw = &lds_x[wave][0];
        #pragma unroll 4
        for (int kk = 0; kk < NKSTEP; ++kk) {
            const int r0 = kk * 4 + half * 2;          // A: K rows owned by this lane half
            v2f a  = *(const v2f*)(xw + nl * XPITCH + r0);
            const float* wb = lds_wB + kk * 256 + lane * 2;
            v2f b0 = *(const v2f*)(wb + 0);            // expert tile 0
            v2f b1 = *(const v2f*)(wb + 64);           // expert tile 1
            v2f b2 = *(const v2f*)(wb + 128);          // expert tile 2
            v2f b3 = *(const v2f*)(wb + 192);          // expert tile 3
            acc0 = __builtin_amdgcn_wmma_f32_16x16x4_f32(false, a, false, b0, (short)0, acc0, false, false);
            acc1 = __builtin_amdgcn_wmma_f32_16x16x4_f32(false, a, false, b1, (short)0, acc1, false, false);
            acc2 = __builtin_amdgcn_wmma_f32_16x16x4_f32(false, a, false, b2, (short)0, acc2, false, false);
            acc3 = __builtin_amdgcn_wmma_f32_16x16x4_f32(false, a, false, b3, (short)0, acc3, false, false);
        }
        __syncthreads();
    }

    // --- per-token scale s_t = rsqrt(mean(x^2)+eps) * H^-1/2 (wave-private) ---
    lds_sq[tid] = sq;
    __builtin_amdgcn_wave_barrier();                   // LDS is in-order within a wave
    float sj[8];
    #pragma unroll
    for (int j = 0; j < 8; ++j) {
        int m = j + half * 8;
        float ss = lds_sq[wave * 32 + 2 * m] + lds_sq[wave * 32 + 2 * m + 1];
        sj[j] = rsqrtf(ss * (1.0f / H_DIM) + 1e-6f) * 0.022097086912079608f; // * H^-0.5
    }

    // --- scaled logits -> LDS (reuse x tile region), C layout: M=j (+8 for high half) ---
    float* lg = &lds_x[wave][0];
    #pragma unroll
    for (int j = 0; j < 8; ++j) {
        int m = j + half * 8;
        lg[m * XPITCH +  0 + nl] = acc0[j] * sj[j];
        lg[m * XPITCH + 16 + nl] = acc1[j] * sj[j];
        lg[m * XPITCH + 32 + nl] = acc2[j] * sj[j];
        lg[m * XPITCH + 48 + nl] = acc3[j] * sj[j];
    }
    __builtin_amdgcn_wave_barrier();

    // --- top-2 over 64 logits per token; softmax denom cancels in renorm ---
    if (lane < 16) {
        const float* row = &lg[lane * XPITCH];
        float b1v = -__builtin_inff(), b2v = -__builtin_inff();
        int   i1 = 0, i2 = 0;
        for (int e = 0; e < E_DIM; ++e) {
            float v = row[e];
            if (v > b1v)      { b2v = b1v; i2 = i1; b1v = v; i1 = e; }
            else if (v > b2v) { b2v = v;   i2 = e; }
        }
        float r  = __expf(b2v - b1v);                  // l2 <= l1 -> no overflow
        float w1 = 1.0f / (1.0f + r);
        float w2 = 1.0f - w1;
        float* wr = &lds_x[wave][lane * XPITCH];
        #pragma unroll
        for (int j4 = 0; j4 < 16; ++j4)
            *(float4*)(wr + j4 * 4) = float4{0.f, 0.f, 0.f, 0.f};
        wr[i1] = w1;
        wr[i2] = w2;
    }
    __builtin_amdgcn_wave_barrier();

    // --- coalesced float4 store of sparse dispatch rows ---
    float*       orow = out + (size_t)(waveTok + mytok) * E_DIM + sub * 32;
    const float* lr   = &lds_x[wave][mytok * XPITCH + sub * 32];
    #pragma unroll
    for (int j = 0; j < 8; ++j)
        *(float4*)(orow + j * 4) = *(const float4*)(lr + j * 4);
}

extern "C" void kernel_launch(void* const* d_in, const int* in_sizes, int n_in,
                              void* d_out, int out_size, void* d_ws, size_t ws_size,
                              hipStream_t stream) {
    const float* x   = (const float*)d_in[0];   // [T, H] fp32
    const float* pds = (const float*)d_in[1];   // [H]    fp32
    const float* W   = (const float*)d_in[2];   // [H, E] fp32
    float* out = (float*)d_out;                 // [T, E] fp32
    float* Wp  = (float*)d_ws;                  // folded W', 512 KB

    fold_scale_kernel<<<(H_DIM * E_DIM) / 256, 256, 0, stream>>>(W, pds, Wp);
    router_kernel<<<T_TOK / TPB, 256, 0, stream>>>(x, Wp, out);
}